// GAT_38560216383776
// MI455X (gfx1250) — compile-verified
//
#include <hip/hip_runtime.h>

// ---------------------------------------------------------------------------
// 4-layer GAT for gfx1250 (MI455X).
//  - Dense GEMM (h @ W) via V_WMMA_F32_16X16X4_F32, one wave per 16x16 tile.
//  - Edge softmax via order-encoded uint atomicMax + float atomicAdd.
//  - Edge aggregation: one wave per edge, float4 coalesced gather + atomics.
// ---------------------------------------------------------------------------

typedef __attribute__((ext_vector_type(2))) float v2f;
typedef __attribute__((ext_vector_type(8))) float v8f;

#define NEG_SLOPE 0.2f

// ---- order-preserving float <-> uint encoding for atomic max ---------------
__device__ __forceinline__ unsigned f2ord(float f) {
  unsigned u = __float_as_uint(f);
  return (u & 0x80000000u) ? ~u : (u | 0x80000000u);
}
__device__ __forceinline__ float ord2f(unsigned u) {
  return (u & 0x80000000u) ? __uint_as_float(u & 0x7fffffffu)
                           : __uint_as_float(~u);
}

// ---------------------------------------------------------------------------
// GEMM: Y[M,Nd] = act(X[M,K]) @ W[K,Nd], act = relu if relu_in else identity.
// One wave computes one 16x16 tile of Y, accumulating K in steps of 4 with
// V_WMMA_F32_16X16X4_F32.
//
// f32 WMMA fragment layout (ISA 7.12.2):
//   A (16x4):  lanes 0-15 hold M=lane, lanes 16-31 hold M=lane-16.
//              VGPR0 = K=0 (lo half) / K=2 (hi half); VGPR1 = K=1 / K=3.
//   B (4x16):  row striped across lanes: VGPR0 = K=0 (lanes 0-15, N=lane)
//              / K=2 (lanes 16-31); VGPR1 = K=1 / K=3.
//   C (16x16): VGPR r = row M=r (lanes 0-15) / M=r+8 (lanes 16-31), N=lane%16.
// ---------------------------------------------------------------------------
__global__ void gat_gemm_f32_wmma(const float* __restrict__ X,
                                  const float* __restrict__ W,
                                  float* __restrict__ Y,
                                  int M, int K, int Nd, int relu_in) {
  const int lane = threadIdx.x & 31;
  const int wavesPerBlk = blockDim.x >> 5;
  const int wave = blockIdx.x * wavesPerBlk + (threadIdx.x >> 5);
  const int ntiles = Nd >> 4;
  const int mtiles = (M + 15) >> 4;
  if (wave >= mtiles * ntiles) return;   // whole-wave exit: EXEC stays all-1s

  const int mt = wave / ntiles;
  const int nt = wave - mt * ntiles;
  const int m0 = mt << 4;
  const int n0 = nt << 4;
  const int half = lane >> 4;            // 0: lanes 0-15, 1: lanes 16-31
  const int l16  = lane & 15;

  int arow = m0 + l16;
  if (arow >= M) arow = M - 1;           // clamp (kept full EXEC); store guarded
  const float* xrow = X + (size_t)arow * K;
  const int bcol = n0 + l16;

  v8f acc = {};
  for (int k0 = 0; k0 < K; k0 += 4) {
    const int ka = k0 + 2 * half;
    float a0 = xrow[ka];
    float a1 = xrow[ka + 1];
    if (relu_in) { a0 = fmaxf(a0, 0.0f); a1 = fmaxf(a1, 0.0f); }
    v2f a; a.x = a0; a.y = a1;
    v2f b;
    b.x = W[(size_t)ka * Nd + bcol];
    b.y = W[(size_t)(ka + 1) * Nd + bcol];
    acc = __builtin_amdgcn_wmma_f32_16x16x4_f32(
        /*neg_a=*/false, a, /*neg_b=*/false, b,
        /*c_mod=*/(short)0, acc, /*reuse_a=*/false, /*reuse_b=*/false);
  }

#pragma unroll
  for (int r = 0; r < 8; ++r) {
    const int row = m0 + r + 8 * half;
    if (row < M) Y[(size_t)row * Nd + bcol] = acc[r];
  }
}

// ---------------------------------------------------------------------------
// Per-node attention scores: s_src[n] = xw[n,:].a_src ; s_dst[n] = xw[n,:].a_dst
// One wave per node, lane-parallel over D, butterfly reduction.
// ---------------------------------------------------------------------------
__global__ void gat_scores(const float* __restrict__ xw,
                           const float* __restrict__ a_src,
                           const float* __restrict__ a_dst,
                           float* __restrict__ s_src,
                           float* __restrict__ s_dst,
                           int Nn, int D) {
  const int wave = (int)((blockIdx.x * blockDim.x + threadIdx.x) >> 5);
  const int lane = threadIdx.x & 31;
  if (wave >= Nn) return;
  const float* row = xw + (size_t)wave * D;
  float ss = 0.0f, sd = 0.0f;
  for (int j = lane; j < D; j += 32) {
    const float v = row[j];
    ss += v * a_src[j];
    sd += v * a_dst[j];
  }
#pragma unroll
  for (int off = 16; off > 0; off >>= 1) {
    ss += __shfl_xor(ss, off, 32);
    sd += __shfl_xor(sd, off, 32);
  }
  if (lane == 0) { s_src[wave] = ss; s_dst[wave] = sd; }
}

// ---- per-layer init --------------------------------------------------------
__global__ void gat_init_node(unsigned* __restrict__ m_enc,
                              float* __restrict__ denom, int Nn) {
  const int t = blockIdx.x * blockDim.x + threadIdx.x;
  if (t < Nn) { m_enc[t] = 0u; denom[t] = 0.0f; }   // 0u encodes <= -inf
}

__global__ void gat_init_bias(float* __restrict__ out,
                              const float* __restrict__ b, int total, int D) {
  const int t = blockIdx.x * blockDim.x + threadIdx.x;
  if (t < total) out[t] = b[t % D];
}

// ---------------------------------------------------------------------------
// Edge pass A: score = leaky_relu(s_src[src] + s_dst[dst]); segment max by dst.
// Edges e < E come from edge_index (int64: row 0 = src, row 1 = dst);
// e >= E are self loops (src = dst = e - E).
// ---------------------------------------------------------------------------
__global__ void gat_edge_max(const long long* __restrict__ ei, int E, int ET,
                             const float* __restrict__ s_src,
                             const float* __restrict__ s_dst,
                             float* __restrict__ score,
                             unsigned* __restrict__ m_enc) {
  const int e = blockIdx.x * blockDim.x + threadIdx.x;
  if (e >= ET) return;
  int s, d;
  if (e < E) { s = (int)ei[e]; d = (int)ei[E + e]; }
  else       { s = d = e - E; }
  float v = s_src[s] + s_dst[d];
  v = (v >= 0.0f) ? v : NEG_SLOPE * v;
  score[e] = v;
  atomicMax(&m_enc[d], f2ord(v));
}

// Edge pass B: ex = exp(score - max[dst]); denom[dst] += ex; score := ex.
__global__ void gat_edge_exp(const long long* __restrict__ ei, int E, int ET,
                             const unsigned* __restrict__ m_enc,
                             float* __restrict__ score,
                             float* __restrict__ denom) {
  const int e = blockIdx.x * blockDim.x + threadIdx.x;
  if (e >= ET) return;
  const int d = (e < E) ? (int)ei[E + e] : (e - E);
  const float ex = expf(score[e] - ord2f(m_enc[d]));
  score[e] = ex;
  atomicAdd(&denom[d], ex);
}

// Edge pass C: out[dst,:] += (ex/denom[dst]) * xw[src,:].
// Thread t handles (edge = t / (D/4), float4-chunk = t % (D/4)); for D=128 one
// wave covers exactly one edge with a coalesced 512B row read.
__global__ void gat_edge_agg(const long long* __restrict__ ei, int E, int ET,
                             const float* __restrict__ ex,
                             const float* __restrict__ denom,
                             const float* __restrict__ xw,
                             float* __restrict__ out, int D) {
  const int chunks = D >> 2;
  const long long tid = (long long)blockIdx.x * blockDim.x + threadIdx.x;
  if (tid >= (long long)ET * chunks) return;
  const int e = (int)(tid / chunks);
  const int c = (int)(tid - (long long)e * chunks);
  int s, d;
  if (e < E) { s = (int)ei[e]; d = (int)ei[E + e]; }
  else       { s = d = e - E; }
  const float alpha = ex[e] / denom[d];
  const float4 v = *(const float4*)(xw + (size_t)s * D + (c << 2));
  float* o = out + (size_t)d * D + (c << 2);
  atomicAdd(o + 0, alpha * v.x);
  atomicAdd(o + 1, alpha * v.y);
  atomicAdd(o + 2, alpha * v.z);
  atomicAdd(o + 3, alpha * v.w);
}

// ---------------------------------------------------------------------------
// Host-side per-layer driver (all launches on `stream`; graph-capture safe).
// ---------------------------------------------------------------------------
static void run_layer(const float* h_in, int relu_in,
                      const float* W, const float* a_s, const float* a_d,
                      const float* b,
                      float* xw, float* out,
                      int Nn, int K, int Dout,
                      const long long* ei, int E, int ET,
                      float* s_src, float* s_dst,
                      unsigned* m_enc, float* denom, float* score,
                      hipStream_t stream) {
  const int mtiles = (Nn + 15) / 16;
  const int ntiles = Dout / 16;
  const int waves = mtiles * ntiles;
  const int wpb = 4;  // 128-thread blocks = 4 waves
  gat_gemm_f32_wmma<<<(waves + wpb - 1) / wpb, wpb * 32, 0, stream>>>(
      h_in, W, xw, Nn, K, Dout, relu_in);

  gat_scores<<<((Nn * 32) + 255) / 256, 256, 0, stream>>>(
      xw, a_s, a_d, s_src, s_dst, Nn, Dout);

  gat_init_node<<<(Nn + 255) / 256, 256, 0, stream>>>(m_enc, denom, Nn);
  gat_init_bias<<<((Nn * Dout) + 255) / 256, 256, 0, stream>>>(out, b,
                                                               Nn * Dout, Dout);

  gat_edge_max<<<(ET + 255) / 256, 256, 0, stream>>>(ei, E, ET, s_src, s_dst,
                                                     score, m_enc);
  gat_edge_exp<<<(ET + 255) / 256, 256, 0, stream>>>(ei, E, ET, m_enc, score,
                                                     denom);
  const long long total = (long long)ET * (Dout >> 2);
  gat_edge_agg<<<(unsigned)((total + 255) / 256), 256, 0, stream>>>(
      ei, E, ET, score, denom, xw, out, Dout);
}

extern "C" void kernel_launch(void* const* d_in, const int* in_sizes, int n_in,
                              void* d_out, int out_size, void* d_ws,
                              size_t ws_size, hipStream_t stream) {
  const int D_IN = 128, D_HID = 128, N_CLS = 16;
  const int Nn = in_sizes[0] / D_IN;       // 50000
  const int E  = in_sizes[1] / 2;          // 800000
  const int ET = E + Nn;                   // + self loops

  const float*     x  = (const float*)d_in[0];
  const long long* ei = (const long long*)d_in[1];
  const float* W1 = (const float*)d_in[2],  *as1 = (const float*)d_in[3],
             *ad1 = (const float*)d_in[4],  *b1  = (const float*)d_in[5];
  const float* W2 = (const float*)d_in[6],  *as2 = (const float*)d_in[7],
             *ad2 = (const float*)d_in[8],  *b2  = (const float*)d_in[9];
  const float* W3 = (const float*)d_in[10], *as3 = (const float*)d_in[11],
             *ad3 = (const float*)d_in[12], *b3  = (const float*)d_in[13];
  const float* W4 = (const float*)d_in[14], *as4 = (const float*)d_in[15],
             *ad4 = (const float*)d_in[16], *b4  = (const float*)d_in[17];

  // ---- workspace carve-out (256B aligned) ----
  char* ws = (char*)d_ws;
  size_t off = 0;
  auto alloc = [&](size_t bytes) {
    void* p = ws + off;
    off += (bytes + 255) & ~(size_t)255;
    return p;
  };
  float*    xw    = (float*)alloc((size_t)Nn * D_HID * sizeof(float));
  float*    hA    = (float*)alloc((size_t)Nn * D_HID * sizeof(float));
  float*    hB    = (float*)alloc((size_t)Nn * D_HID * sizeof(float));
  float*    s_src = (float*)alloc((size_t)Nn * sizeof(float));
  float*    s_dst = (float*)alloc((size_t)Nn * sizeof(float));
  unsigned* m_enc = (unsigned*)alloc((size_t)Nn * sizeof(unsigned));
  float*    denom = (float*)alloc((size_t)Nn * sizeof(float));
  float*    score = (float*)alloc((size_t)ET * sizeof(float));
  (void)ws_size; (void)n_in; (void)out_size;

  // Layer 1: x -> hA (relu deferred: fused into next GEMM's A load)
  run_layer(x,  0, W1, as1, ad1, b1, xw, hA, Nn, D_IN,  D_HID, ei, E, ET,
            s_src, s_dst, m_enc, denom, score, stream);
  // Layer 2: relu(hA) -> hB
  run_layer(hA, 1, W2, as2, ad2, b2, xw, hB, Nn, D_HID, D_HID, ei, E, ET,
            s_src, s_dst, m_enc, denom, score, stream);
  // Layer 3: relu(hB) -> hA
  run_layer(hB, 1, W3, as3, ad3, b3, xw, hA, Nn, D_HID, D_HID, ei, E, ET,
            s_src, s_dst, m_enc, denom, score, stream);
  // Layer 4: relu(hA) -> d_out (no trailing relu)
  run_layer(hA, 1, W4, as4, ad4, b4, xw, (float*)d_out, Nn, D_HID, N_CLS,
            ei, E, ET, s_src, s_dst, m_enc, denom, score, stream);
}